// Performer_5815385718824
// MI455X (gfx1250) — compile-verified
//
#include <hip/hip_runtime.h>
#include <cmath>

// ---------------------------------------------------------------------------
// Performer FAVOR+ attention for MI455X (gfx1250, wave32, WMMA f32 16x16x4).
// BH=64, N=8192, D=64, M=32, all fp32. Bandwidth-bound (>=512MB moved,
// ~8.6 GFLOP) -> keep f32 precision, use V_WMMA_F32_16X16X4_F32 for all GEMMs.
// ---------------------------------------------------------------------------

#define BH 64
#define NSEQ 8192
#define DDIM 64
#define MFEAT 32
#define SSPLIT 8            // N-splits for the kv accumulation pass

#define NRM   0.3535533905932738f   // 64^-0.25
#define RATIO 0.17677669529663687f  // 32^-0.5
#define DIAGC 0.0625f               // 0.5 * NRM^2
#define EPSK  1e-4f

typedef float v2f __attribute__((ext_vector_type(2)));
typedef float v8f __attribute__((ext_vector_type(8)));

#if defined(__HIP_DEVICE_COMPILE__)
#if !__has_builtin(__builtin_amdgcn_wmma_f32_16x16x4_f32)
#error "device pass: missing __builtin_amdgcn_wmma_f32_16x16x4_f32"
#endif
#endif

__device__ __forceinline__ v8f wmma4(v2f a, v2f b, v8f c) {
#if defined(__HIP_DEVICE_COMPILE__)
  // D = A(16x4) * B(4x16) + C(16x16), all f32, wave32
  return __builtin_amdgcn_wmma_f32_16x16x4_f32(false, a, false, b, (short)0, c,
                                               false, false);
#else
  (void)a; (void)b;
  return c;  // host pass stub; never executed
#endif
}

__device__ __forceinline__ int f2ord(float f) {
  int b = __float_as_int(f);
  return b >= 0 ? b : (b ^ 0x7fffffff);
}
__device__ __forceinline__ float ord2f(int k) {
  return __int_as_float(k >= 0 ? k : (k ^ 0x7fffffff));
}

// Load proj^T as WMMA B-operand register tiles.
// bq[mt][kk]: lane l<16 holds B[4kk+0..1][mt*16+l], lane l+16 holds B[4kk+2..3][..]
__device__ __forceinline__ void load_projB(const float* __restrict__ proj,
                                           int l15, int hi, v2f bq0[16],
                                           v2f bq1[16]) {
#pragma unroll
  for (int kk = 0; kk < 16; ++kk) {
    bq0[kk] = *(const v2f*)(proj + (0 * 16 + l15) * DDIM + kk * 4 + 2 * hi);
    bq1[kk] = *(const v2f*)(proj + (1 * 16 + l15) * DDIM + kk * 4 + 2 * hi);
  }
}

// Compute a [16 x 32] logits tile:  (NRM*data[rows]) @ proj^T, plus diag term.
// base points at data[bh][row0][0].  c0/c1 are the two 16x16 C tiles,
// diagv at lane r holds diag of local row (r&15).
__device__ __forceinline__ void logits_tile(const float* __restrict__ base,
                                            const v2f bq0[16],
                                            const v2f bq1[16], int l15, int hi,
                                            v8f& c0, v8f& c1, float& diagv) {
  v8f z = {0.f, 0.f, 0.f, 0.f, 0.f, 0.f, 0.f, 0.f};
  c0 = z;
  c1 = z;
  float ss = 0.f;
  const float* rp = base + (size_t)l15 * DDIM + 2 * hi;
#pragma unroll
  for (int kk = 0; kk < 16; ++kk) {
    v2f a = *(const v2f*)(rp + 4 * kk);
    ss += a.x * a.x + a.y * a.y;
    v2f an;
    an.x = a.x * NRM;
    an.y = a.y * NRM;
    c0 = wmma4(an, bq0[kk], c0);
    c1 = wmma4(an, bq1[kk], c1);
  }
  diagv = DIAGC * (ss + __shfl_xor(ss, 16, 32));
}

// --------------------------- kernel 0: init max ----------------------------
__global__ void perf_init_kernel(int* gmax) { gmax[0] = (int)0x80000000; }

// ------------------- kernel 1: global max of k @ proj^T --------------------
__global__ void perf_kmax_kernel(const float* __restrict__ k,
                                 const float* __restrict__ proj,
                                 int* __restrict__ gmax) {
  const int tid = threadIdx.x;
  const int lane = tid & 31, w = tid >> 5;
  const int l15 = lane & 15, hi = lane >> 4;
  const int bh = blockIdx.x >> 6;          // N/128 = 64 chunks per bh
  const int chunk = blockIdx.x & 63;
  const int row0 = chunk * 128 + w * 16;

  v2f bq0[16], bq1[16];
  load_projB(proj, l15, hi, bq0, bq1);

  v8f c0, c1;
  float diagv;
  const float* base = k + ((size_t)bh * NSEQ + row0) * DDIM;
  logits_tile(base, bq0, bq1, l15, hi, c0, c1, diagv);

  float mx = -INFINITY;
#pragma unroll
  for (int j = 0; j < 8; ++j) mx = fmaxf(mx, fmaxf(c0[j], c1[j]));
#pragma unroll
  for (int off = 1; off < 32; off <<= 1) mx = fmaxf(mx, __shfl_xor(mx, off, 32));

  __shared__ float smax[8];
  if (lane == 0) smax[w] = mx;
  __syncthreads();
  if (tid == 0) {
    float m = smax[0];
#pragma unroll
    for (int i = 1; i < 8; ++i) m = fmaxf(m, smax[i]);
    atomicMax(gmax, f2ord(m));
  }
}

// -------------- kernel 2: kp -> kv = kp^T @ v, ksum = sum_n kp -------------
__global__ void perf_kv_kernel(const float* __restrict__ k,
                               const float* __restrict__ v,
                               const float* __restrict__ proj,
                               const int* __restrict__ gmaxi,
                               float* __restrict__ kvp,
                               float* __restrict__ ksp) {
  const int tid = threadIdx.x;
  const int lane = tid & 31, w = tid >> 5;
  const int l15 = lane & 15, hi = lane >> 4;
  const int bh = blockIdx.x / SSPLIT;
  const int s = blockIdx.x % SSPLIT;
  const int nper = NSEQ / SSPLIT;  // 1024 rows per block

  __shared__ float kpbuf[128][33];   // [n][m], padded (stride 33 coprime w/ 64)
  __shared__ float vbuf[128][72];    // [n][d], pad 8 -> halves hit disjoint banks
  __shared__ float ksacc[MFEAT];

  v2f bq0[16], bq1[16];
  load_projB(proj, l15, hi, bq0, bq1);
  const float gmax = ord2f(gmaxi[0]);

  if (tid < MFEAT) ksacc[tid] = 0.f;

  const int mt = w >> 2, dt = w & 3;
  const int m = mt * 16 + l15, d = dt * 16 + l15;
  v8f acc = {0.f, 0.f, 0.f, 0.f, 0.f, 0.f, 0.f, 0.f};
  float ks0 = 0.f, ks1 = 0.f;

  const float* kbase = k + (size_t)bh * NSEQ * DDIM;
  const float* vbase = v + (size_t)bh * NSEQ * DDIM;

  for (int it = 0; it < nper / 128; ++it) {
    const int n0 = s * nper + it * 128;

    // phase A: this wave computes kp for 16 rows, stores to LDS
    v8f c0, c1;
    float diagv;
    logits_tile(kbase + (size_t)(n0 + w * 16) * DDIM, bq0, bq1, l15, hi, c0, c1,
                diagv);
#pragma unroll
    for (int j = 0; j < 8; ++j) {
      const float dj = __shfl(diagv, j + 8 * hi, 32);
      const float p0 = RATIO * (__expf(c0[j] - dj - gmax) + EPSK);
      const float p1 = RATIO * (__expf(c1[j] - dj - gmax) + EPSK);
      kpbuf[w * 16 + j + 8 * hi][l15] = p0;
      kpbuf[w * 16 + j + 8 * hi][16 + l15] = p1;
      ks0 += p0;
      ks1 += p1;
    }
    // cooperative coalesced load of v tile [128 x 64] into LDS
    for (int t = tid; t < 128 * 16; t += 256) {
      const int r = t >> 4, cq = t & 15;
      const float4 val =
          *(const float4*)(vbase + (size_t)(n0 + r) * DDIM + cq * 4);
      *(float4*)(&vbuf[r][cq * 4]) = val;
    }
    __syncthreads();

    // phase B: each wave owns kv tile (mt,dt); A = kp^T from LDS, B = v
#pragma unroll 8
    for (int nk = 0; nk < 128; nk += 4) {
      v2f a, b;
      a.x = kpbuf[nk + 2 * hi][m];
      a.y = kpbuf[nk + 2 * hi + 1][m];
      b.x = vbuf[nk + 2 * hi][d];
      b.y = vbuf[nk + 2 * hi + 1][d];
      acc = wmma4(a, b, acc);
    }
    __syncthreads();
  }

  // ksum: combine halves (lanes l and l+16 hold same m), accumulate in LDS
  const float k0c = ks0 + __shfl_xor(ks0, 16, 32);
  const float k1c = ks1 + __shfl_xor(ks1, 16, 32);
  if (lane < 16) {
    atomicAdd(&ksacc[l15], k0c);
    atomicAdd(&ksacc[16 + l15], k1c);
  }
  __syncthreads();
  const size_t pbase = (size_t)(bh * SSPLIT + s);
  if (tid < MFEAT) ksp[pbase * MFEAT + tid] = ksacc[tid];
#pragma unroll
  for (int j = 0; j < 8; ++j) {
    const int mrow = mt * 16 + j + 8 * hi;
    kvp[(pbase * MFEAT + mrow) * DDIM + d] = acc[j];
  }
}

// ----------------- kernel 3: reduce the SSPLIT partials --------------------
__global__ void perf_reduce_kernel(const float* __restrict__ kvp,
                                   const float* __restrict__ ksp,
                                   float* __restrict__ kvf,
                                   float* __restrict__ ksf) {
  const int idx = blockIdx.x * 256 + threadIdx.x;
  const int NKV = BH * MFEAT * DDIM;  // 131072
  if (idx < NKV) {
    const int bh = idx / (MFEAT * DDIM);
    const int r = idx % (MFEAT * DDIM);
    float sum = 0.f;
#pragma unroll
    for (int s = 0; s < SSPLIT; ++s)
      sum += kvp[((size_t)(bh * SSPLIT + s)) * (MFEAT * DDIM) + r];
    kvf[idx] = sum;
  } else if (idx < NKV + BH * MFEAT) {
    const int i2 = idx - NKV;
    const int bh = i2 / MFEAT, mm = i2 % MFEAT;
    float sum = 0.f;
#pragma unroll
    for (int s = 0; s < SSPLIT; ++s)
      sum += ksp[(size_t)(bh * SSPLIT + s) * MFEAT + mm];
    ksf[i2] = sum;
  }
}

// ------------- kernel 4: qp, num = qp@kv, den = qp@ksum, out ---------------
__global__ void perf_out_kernel(const float* __restrict__ q,
                                const float* __restrict__ proj,
                                const float* __restrict__ kvf,
                                const float* __restrict__ ksf,
                                float* __restrict__ out) {
  const int tid = threadIdx.x;
  const int lane = tid & 31, w = tid >> 5;
  const int l15 = lane & 15, hi = lane >> 4;
  const int bh = blockIdx.x >> 6;
  const int chunk = blockIdx.x & 63;
  const int row0 = chunk * 128;

  __shared__ float qpbuf[128][33];
  __shared__ float kvbuf[MFEAT][72];
  __shared__ float ksbuf[MFEAT];
  __shared__ float denbuf[128];

  v2f bq0[16], bq1[16];
  load_projB(proj, l15, hi, bq0, bq1);

  for (int t = tid; t < MFEAT * DDIM; t += 256)
    kvbuf[t >> 6][t & 63] = kvf[(size_t)bh * MFEAT * DDIM + t];
  if (tid < MFEAT) ksbuf[tid] = ksf[bh * MFEAT + tid];
  __syncthreads();

  // qp for this wave's 16 rows (per-row max, diag, exp)
  v8f c0, c1;
  float diagv;
  const float* base = q + ((size_t)bh * NSEQ + row0 + w * 16) * DDIM;
  logits_tile(base, bq0, bq1, l15, hi, c0, c1, diagv);
#pragma unroll
  for (int j = 0; j < 8; ++j) {
    float mj = fmaxf(c0[j], c1[j]);  // row (j + 8*hi) spread over 16 lanes
#pragma unroll
    for (int off = 1; off < 16; off <<= 1) mj = fmaxf(mj, __shfl_xor(mj, off, 32));
    const float dj = __shfl(diagv, j + 8 * hi, 32);
    const float p0 = RATIO * (__expf(c0[j] - dj - mj) + EPSK);
    const float p1 = RATIO * (__expf(c1[j] - dj - mj) + EPSK);
    qpbuf[w * 16 + j + 8 * hi][l15] = p0;
    qpbuf[w * 16 + j + 8 * hi][16 + l15] = p1;
  }
  __syncthreads();

  // den[row] = qp[row] . ksum   (lane < 16 each owns one of the wave's rows)
  if (lane < 16) {
    const int row = w * 16 + l15;
    float den = 0.f;
#pragma unroll
    for (int mm = 0; mm < MFEAT; ++mm) den += qpbuf[row][mm] * ksbuf[mm];
    denbuf[row] = den;
  }
  __syncthreads();

  // num = qp(16x32) @ kv(32x64) per wave, then divide and store
#pragma unroll
  for (int dt = 0; dt < 4; ++dt) {
    v8f c = {0.f, 0.f, 0.f, 0.f, 0.f, 0.f, 0.f, 0.f};
#pragma unroll
    for (int kk = 0; kk < 8; ++kk) {
      const int k0 = 4 * kk;
      v2f a, b;
      a.x = qpbuf[w * 16 + l15][k0 + 2 * hi];
      a.y = qpbuf[w * 16 + l15][k0 + 2 * hi + 1];
      b.x = kvbuf[k0 + 2 * hi][dt * 16 + l15];
      b.y = kvbuf[k0 + 2 * hi + 1][dt * 16 + l15];
      c = wmma4(a, b, c);
    }
#pragma unroll
    for (int j = 0; j < 8; ++j) {
      const int rloc = w * 16 + j + 8 * hi;
      const float dn = denbuf[rloc];
      out[((size_t)bh * NSEQ + row0 + rloc) * DDIM + dt * 16 + l15] = c[j] / dn;
    }
  }
}

// ---------------------------------------------------------------------------
extern "C" void kernel_launch(void* const* d_in, const int* in_sizes, int n_in,
                              void* d_out, int out_size, void* d_ws,
                              size_t ws_size, hipStream_t stream) {
  const float* q = (const float*)d_in[0];
  const float* k = (const float*)d_in[1];
  const float* v = (const float*)d_in[2];
  const float* proj = (const float*)d_in[3];
  float* out = (float*)d_out;

  int* gmax = (int*)d_ws;
  float* kvp = (float*)((char*)d_ws + 256);
  float* ksp = kvp + (size_t)BH * SSPLIT * MFEAT * DDIM;  // 4 MB of partial kv
  float* kvf = ksp + (size_t)BH * SSPLIT * MFEAT;
  float* ksf = kvf + (size_t)BH * MFEAT * DDIM;

  perf_init_kernel<<<1, 1, 0, stream>>>(gmax);
  perf_kmax_kernel<<<BH * (NSEQ / 128), 256, 0, stream>>>(k, proj, gmax);
  perf_kv_kernel<<<BH * SSPLIT, 256, 0, stream>>>(k, v, proj, gmax, kvp, ksp);
  perf_reduce_kernel<<<(BH * MFEAT * DDIM + BH * MFEAT + 255) / 256, 256, 0,
                       stream>>>(kvp, ksp, kvf, ksf);
  perf_out_kernel<<<BH * (NSEQ / 128), 256, 0, stream>>>(q, proj, kvf, ksf,
                                                         out);
}